// GATConv_6932077216253
// MI455X (gfx1250) — compile-verified
//
#include <hip/hip_runtime.h>

typedef float v2f __attribute__((ext_vector_type(2)));
typedef float v8f __attribute__((ext_vector_type(8)));

#define N_NODES   50000
#define N_EDGES   800000
#define D_IN      256
#define N_HEADS   8
#define D_OUT     32
#define D_TOT     256       // N_HEADS * D_OUT
#define NEG_SLOPE 0.2f
#define ROW_TILES (N_NODES / 16)   // 3125 (exact)

// ---- order-preserving float <-> uint key (for atomicMax edge-softmax max) ----
__device__ __forceinline__ unsigned fkey(float f) {
  unsigned u = __float_as_uint(f);
  return (u & 0x80000000u) ? ~u : (u | 0x80000000u);
}
__device__ __forceinline__ float fdec(unsigned k) {
  return (k & 0x80000000u) ? __uint_as_float(k & 0x7fffffffu)
                           : __uint_as_float(~k);
}

// =====================================================================
// Kernel 1: proj = feat @ W  via V_WMMA_F32_16X16X4_F32
//   Block = 8 waves = 8 row-tiles x one 64-col group.
//   The 256x64 W slice (64 KB) is staged once per block into LDS,
//   pair-interleaved so each B fragment is one ds_load_b64.
// =====================================================================
__global__ __launch_bounds__(256) void gat_gemm_wmma(
    const float* __restrict__ A,   // [N_NODES, D_IN]
    const float* __restrict__ B,   // [D_IN, D_TOT]
    float* __restrict__ C)         // [N_NODES, D_TOT]
{
  // bs[(k>>1)*128 + c*2 + (k&1)] = W[k][colGrp*64 + c]
  __shared__ float bs[D_IN * 64];            // 64 KB of the 320 KB WGP LDS

  const int tid       = threadIdx.x;
  const int lane      = tid & 31;
  const int waveInBlk = tid >> 5;
  const int rowBlock  = blockIdx.x >> 2;     // 0..390
  const int colGrp    = blockIdx.x & 3;      // 0..3

  // ---- cooperative stage of W[:, colGrp*64 .. +64) ----
  const float* wslice = B + colGrp * 64;
  #pragma unroll 4
  for (int i = 0; i < 64; ++i) {
    int flat = i * 256 + tid;                // 0..16383
    int k = flat >> 6;                       // 0..255
    int c = flat & 63;
    bs[(k >> 1) * 128 + c * 2 + (k & 1)] = wslice[(size_t)k * D_TOT + c];
  }
  __syncthreads();

  const int rowTile = rowBlock * 8 + waveInBlk;
  if (rowTile < ROW_TILES) {                 // wave-uniform guard; EXEC all-1s for WMMA
    const int m     = lane & 15;
    const int khalf = (lane >> 4) << 1;      // 0 or 2
    const int ncol  = lane & 15;
    const float* arow = A + (size_t)(rowTile * 16 + m) * D_IN;
    // per-lane LDS base: column pair offset + which half of the k-quad
    const float* bbase = bs + ncol * 2 + (khalf >> 1) * 128;

    v8f acc0 = {}, acc1 = {}, acc2 = {}, acc3 = {};

    for (int k0 = 0; k0 < D_IN; k0 += 4) {
      // A fragment 16x4: lane m holds K = khalf, khalf+1 (one b64 load)
      v2f a;
      const float* ap = arow + k0 + khalf;
      a.x = ap[0];
      a.y = ap[1];

      const float* bk = bbase + (k0 >> 1) * 128;   // pair row for this k-quad
      v2f b0 = *(const v2f*)(bk + 0 * 32);
      v2f b1 = *(const v2f*)(bk + 1 * 32);
      v2f b2 = *(const v2f*)(bk + 2 * 32);
      v2f b3 = *(const v2f*)(bk + 3 * 32);

      acc0 = __builtin_amdgcn_wmma_f32_16x16x4_f32(false, a, false, b0, (short)0, acc0, false, false);
      acc1 = __builtin_amdgcn_wmma_f32_16x16x4_f32(false, a, false, b1, (short)0, acc1, false, false);
      acc2 = __builtin_amdgcn_wmma_f32_16x16x4_f32(false, a, false, b2, (short)0, acc2, false, false);
      acc3 = __builtin_amdgcn_wmma_f32_16x16x4_f32(false, a, false, b3, (short)0, acc3, false, false);
    }

    // C/D layout: VGPR i -> row i (lanes 0-15) / row i+8 (lanes 16-31); col = lane&15
    const int rbase = rowTile * 16 + ((lane >> 4) << 3);
    #pragma unroll
    for (int i = 0; i < 8; ++i) {
      size_t r = (size_t)(rbase + i) * D_TOT;
      C[r + colGrp * 64 +  0 + ncol] = acc0[i];
      C[r + colGrp * 64 + 16 + ncol] = acc1[i];
      C[r + colGrp * 64 + 32 + ncol] = acc2[i];
      C[r + colGrp * 64 + 48 + ncol] = acc3[i];
    }
  }
}

// =====================================================================
// Kernel 2: per-node attention logits  el[n,h], er[n,h]
// =====================================================================
__global__ __launch_bounds__(256) void gat_logits(
    const float* __restrict__ proj,
    const float* __restrict__ attn_l,   // [H, D_OUT]
    const float* __restrict__ attn_r,
    float* __restrict__ el, float* __restrict__ er)
{
  int idx = blockIdx.x * blockDim.x + threadIdx.x;
  if (idx >= N_NODES * N_HEADS) return;
  int h = idx & 7;
  int n = idx >> 3;
  const float4* p  = (const float4*)(proj + (size_t)n * D_TOT + h * D_OUT);
  const float4* al = (const float4*)(attn_l + h * D_OUT);
  const float4* ar = (const float4*)(attn_r + h * D_OUT);
  float sl = 0.f, sr = 0.f;
  #pragma unroll
  for (int i = 0; i < D_OUT / 4; ++i) {
    float4 v = p[i], l = al[i], r = ar[i];
    sl += v.x * l.x + v.y * l.y + v.z * l.z + v.w * l.w;
    sr += v.x * r.x + v.y * r.y + v.z * r.z + v.w * r.w;
  }
  el[idx] = sl;
  er[idx] = sr;
}

// =====================================================================
// Kernel 3: segment-max of leaky scores into ordered-uint keys
// =====================================================================
__global__ __launch_bounds__(256) void gat_edge_max(
    const int* __restrict__ src, const int* __restrict__ dst,
    const float* __restrict__ el, const float* __restrict__ er,
    unsigned* __restrict__ emaxk)
{
  int e = blockIdx.x * blockDim.x + threadIdx.x;
  if (e >= N_EDGES) return;
  int u = src[e], v = dst[e];
  const float* pl = el + (size_t)u * N_HEADS;
  const float* pr = er + (size_t)v * N_HEADS;
  unsigned* mk = emaxk + (size_t)v * N_HEADS;
  #pragma unroll
  for (int h = 0; h < N_HEADS; ++h) {
    float s = pl[h] + pr[h];
    s = s > 0.f ? s : NEG_SLOPE * s;
    atomicMax(&mk[h], fkey(s));
  }
}

// =====================================================================
// Kernel 4: segment-sum of exp(s - max)
// =====================================================================
__global__ __launch_bounds__(256) void gat_edge_sum(
    const int* __restrict__ src, const int* __restrict__ dst,
    const float* __restrict__ el, const float* __restrict__ er,
    const unsigned* __restrict__ emaxk, float* __restrict__ esum)
{
  int e = blockIdx.x * blockDim.x + threadIdx.x;
  if (e >= N_EDGES) return;
  int u = src[e], v = dst[e];
  const float* pl = el + (size_t)u * N_HEADS;
  const float* pr = er + (size_t)v * N_HEADS;
  const unsigned* mk = emaxk + (size_t)v * N_HEADS;
  float* sm = esum + (size_t)v * N_HEADS;
  #pragma unroll
  for (int h = 0; h < N_HEADS; ++h) {
    float s = pl[h] + pr[h];
    s = s > 0.f ? s : NEG_SLOPE * s;
    float ee = expf(s - fdec(mk[h]));   // dst with >=1 edge always has finite max
    atomicAdd(&sm[h], ee);
  }
}

// =====================================================================
// Kernel 5: weighted scatter  out[dst] += a * proj[src]
//   one wave32 per edge; lane -> 8 contiguous channels, one head each.
// =====================================================================
__global__ __launch_bounds__(256) void gat_scatter(
    const int* __restrict__ src, const int* __restrict__ dst,
    const float* __restrict__ el, const float* __restrict__ er,
    const unsigned* __restrict__ emaxk, const float* __restrict__ esum,
    const float* __restrict__ proj, float* __restrict__ out)
{
  const int lane = threadIdx.x & 31;
  const int e    = blockIdx.x * 8 + (threadIdx.x >> 5);
  if (e >= N_EDGES) return;
  const int u = src[e], v = dst[e];
  const int h = lane >> 2;                 // 32 channels/head, 8 channels/lane

  float s = el[(size_t)u * N_HEADS + h] + er[(size_t)v * N_HEADS + h];
  s = s > 0.f ? s : NEG_SLOPE * s;
  float m  = fdec(emaxk[(size_t)v * N_HEADS + h]);
  float a  = expf(s - m) / esum[(size_t)v * N_HEADS + h];  // esum>0 for any dst with edges

  const int base = lane * 8;
  const float4* p = (const float4*)(proj + (size_t)u * D_TOT + base);
  float4 p0 = p[0], p1 = p[1];
  float* o = out + (size_t)v * D_TOT + base;
  atomicAdd(o + 0, a * p0.x);
  atomicAdd(o + 1, a * p0.y);
  atomicAdd(o + 2, a * p0.z);
  atomicAdd(o + 3, a * p0.w);
  atomicAdd(o + 4, a * p1.x);
  atomicAdd(o + 5, a * p1.y);
  atomicAdd(o + 6, a * p1.z);
  atomicAdd(o + 7, a * p1.w);
}

// =====================================================================
extern "C" void kernel_launch(void* const* d_in, const int* in_sizes, int n_in,
                              void* d_out, int out_size, void* d_ws, size_t ws_size,
                              hipStream_t stream) {
  const float* feat   = (const float*)d_in[0];
  const float* W      = (const float*)d_in[1];
  const float* attn_l = (const float*)d_in[2];
  const float* attn_r = (const float*)d_in[3];
  const int*   src    = (const int*)d_in[4];
  const int*   dst    = (const int*)d_in[5];
  float* out = (float*)d_out;

  // workspace carve-out (~58 MB)
  char* ws = (char*)d_ws;
  size_t off = 0;
  auto carve = [&](size_t bytes) -> void* {
    void* p = ws + off;
    off += (bytes + 255) & ~(size_t)255;
    return p;
  };
  float*    proj  = (float*)   carve(sizeof(float) * (size_t)N_NODES * D_TOT);
  float*    el    = (float*)   carve(sizeof(float) * (size_t)N_NODES * N_HEADS);
  float*    er    = (float*)   carve(sizeof(float) * (size_t)N_NODES * N_HEADS);
  unsigned* emaxk = (unsigned*)carve(sizeof(unsigned) * (size_t)N_NODES * N_HEADS);
  float*    esum  = (float*)   carve(sizeof(float) * (size_t)N_NODES * N_HEADS);
  (void)ws_size; (void)in_sizes; (void)n_in; (void)out_size;

  // zero-init (graph-capture safe): out = 0, emax keys = 0 (decodes non-finite), esum = 0
  hipMemsetAsync(out,   0, sizeof(float)    * (size_t)N_NODES * D_TOT,   stream);
  hipMemsetAsync(emaxk, 0, sizeof(unsigned) * (size_t)N_NODES * N_HEADS, stream);
  hipMemsetAsync(esum,  0, sizeof(float)    * (size_t)N_NODES * N_HEADS, stream);

  // 1) projection GEMM (WMMA f32): 391 row-blocks (8 row-tiles each) x 4 col-groups
  {
    int rowBlocks = (ROW_TILES + 7) / 8;     // 391
    gat_gemm_wmma<<<rowBlocks * 4, 256, 0, stream>>>(feat, W, proj);
  }
  // 2) per-node logits
  gat_logits<<<(N_NODES * N_HEADS + 255) / 256, 256, 0, stream>>>(proj, attn_l, attn_r, el, er);
  // 3) segment max
  gat_edge_max<<<(N_EDGES + 255) / 256, 256, 0, stream>>>(src, dst, el, er, emaxk);
  // 4) segment sum of exp
  gat_edge_sum<<<(N_EDGES + 255) / 256, 256, 0, stream>>>(src, dst, el, er, emaxk, esum);
  // 5) weighted scatter (wave per edge)
  gat_scatter<<<(N_EDGES + 7) / 8, 256, 0, stream>>>(src, dst, el, er, emaxk, esum, proj, out);
}